// GaussRenderer_21234318311912
// MI455X (gfx1250) — compile-verified
//
#include <hip/hip_runtime.h>
#include <hip/hip_bf16.h>
#include <math.h>

// Gaussian splat renderer for MI455X (gfx1250, wave32).
//
// Strategy:
//  - gs_prep: stable O(N^2) depth rank (matches jnp.argsort stable order),
//    2x2 covariance inversion, scatter into depth-sorted SoA table in d_ws.
//  - gs_render: each wave owns 16 pixels of one x-column (consecutive y).
//    Per-tile of 16 sorted Gaussians, the power -0.5 d^T Sigma^-1 d reduces
//    to a per-Gaussian quadratic in t = y - y0, i.e. a K=4 GEMM evaluated
//    with one V_WMMA_F32_16X16X4_F32 (A = Gaussian coeffs, B = [1, t, t^2, 0]).
//    D layout gives each lane 8 consecutive sorted Gaussians for its pixel;
//    the front-to-back transmittance chain across the two half-wave segments
//    is stitched with the segmented identity S = S_lo + Q_lo*S_hi, Q = Q_lo*Q_hi
//    using __shfl_xor(.,16).

#define GS_W 256
#define GS_H 256
#define GS_N 1024

typedef __attribute__((ext_vector_type(2))) float v2f;
typedef __attribute__((ext_vector_type(8))) float v8f;

// ---------------------------------------------------------------------------
// Kernel 1: stable depth sort (rank by counting) + covariance inversion +
// scatter into sorted SoA layout in workspace:
//   ws[0..1023]      mx        ws[5120..] opacity
//   ws[1024..]       my        ws[6144..] color.r
//   ws[2048..]       invA      ws[7168..] color.g
//   ws[3072..]       invB      ws[8192..] color.b
//   ws[4096..]       invC
// ---------------------------------------------------------------------------
__global__ __launch_bounds__(GS_N) void gs_prep(
    const float* __restrict__ means2D, const float* __restrict__ depths,
    const float* __restrict__ cov2D, const float* __restrict__ color,
    const float* __restrict__ opacity, float* __restrict__ ws) {
  __shared__ float sd[GS_N];
  const int i = threadIdx.x;
  sd[i] = depths[i];
  __syncthreads();
  const float di = sd[i];
  int r = 0;
  for (int j = 0; j < GS_N; ++j) {
    const float dj = sd[j];
    r += (dj < di) || (dj == di && j < i);
  }
  const float a = cov2D[4 * i + 0];
  const float b = cov2D[4 * i + 1];
  const float c = cov2D[4 * i + 3];
  const float inv_det = 1.0f / (a * c - b * b);
  ws[r + 0 * GS_N] = means2D[2 * i + 0];
  ws[r + 1 * GS_N] = means2D[2 * i + 1];
  ws[r + 2 * GS_N] = c * inv_det;
  ws[r + 3 * GS_N] = -b * inv_det;
  ws[r + 4 * GS_N] = a * inv_det;
  ws[r + 5 * GS_N] = opacity[i];
  ws[r + 6 * GS_N] = color[3 * i + 0];
  ws[r + 7 * GS_N] = color[3 * i + 1];
  ws[r + 8 * GS_N] = color[3 * i + 2];
}

// ---------------------------------------------------------------------------
// Kernel 2: render. 512 blocks x 256 threads (8 waves). Wave w handles pixels
// p = w*16 .. w*16+15 where p = x*H + y  (x = p>>8, y = p&255): one x-column,
// 16 consecutive y. Whole sorted Gaussian table lives in LDS (36 KB).
// ---------------------------------------------------------------------------
__global__ __launch_bounds__(256) void gs_render(
    const float* __restrict__ gws, const float* __restrict__ bg,
    float* __restrict__ out) {
  __shared__ float lds[9 * GS_N];
  for (int idx = threadIdx.x; idx < 9 * GS_N; idx += 256) lds[idx] = gws[idx];
  __syncthreads();

  const float* mx = lds + 0 * GS_N;
  const float* my = lds + 1 * GS_N;
  const float* ia = lds + 2 * GS_N;
  const float* ib = lds + 3 * GS_N;
  const float* ic = lds + 4 * GS_N;
  const float* op = lds + 5 * GS_N;
  const float* cr = lds + 6 * GS_N;
  const float* cg = lds + 7 * GS_N;
  const float* cb = lds + 8 * GS_N;

  const int lane = threadIdx.x & 31;
  const int warp = threadIdx.x >> 5;
  const int wid = blockIdx.x * 8 + warp;  // 0..4095
  const int lo = lane & 15;
  const bool hi = lane >= 16;

  const int pbase = wid * 16;
  const int x = pbase >> 8;      // column (shared by the wave)
  const int y0 = pbase & 255;    // first row of the wave's 16 pixels
  const float xf = (float)x;
  const float y0f = (float)y0;

  // B matrix (4 x 16 pixels), constant for the whole wave:
  //   VGPR0: K=0 -> 1.0 (lanes 0-15), K=2 -> t^2 (lanes 16-31)
  //   VGPR1: K=1 -> t   (lanes 0-15), K=3 -> 0   (lanes 16-31)
  const float t = (float)lo;
  v2f bv;
  bv.x = hi ? t * t : 1.0f;
  bv.y = hi ? 0.0f : t;

  float T = 1.0f;
  float accR = 0.0f, accG = 0.0f, accB = 0.0f;

  for (int tile = 0; tile < GS_N; tile += 16) {
    // Build A matrix (16 Gaussians x 4 coeffs). Lane pair (lo, lo+16) covers
    // Gaussian tile+lo:  power(t) = g0 + g1*t + g2*t^2 with
    //   dx = mx - x, u = my - y0
    //   g0 = -(0.5*ia*dx^2 + ib*dx*u + 0.5*ic*u^2)
    //   g1 = ib*dx + ic*u
    //   g2 = -0.5*ic
    const int g = tile + lo;
    const float dx = mx[g] - xf;
    const float u = my[g] - y0f;
    const float A_ = ia[g], Bc = ib[g], Cc = ic[g];
    const float g0 = -(0.5f * A_ * dx * dx + Bc * dx * u + 0.5f * Cc * u * u);
    const float g1 = Bc * dx + Cc * u;
    v2f av;
    av.x = hi ? (-0.5f * Cc) : g0;
    av.y = hi ? 0.0f : g1;

    // power tile: D[M=gaussian, N=pixel], lane holds 8 Gaussians for pixel lo.
    v8f zero = {};
    v8f pw = __builtin_amdgcn_wmma_f32_16x16x4_f32(
        /*neg_a=*/false, av, /*neg_b=*/false, bv,
        /*c_mod=*/(short)0, zero, /*reuse_a=*/false, /*reuse_b=*/false);

    // Local front-to-back scan over this lane's 8 Gaussians (local T=1).
    float Sr = 0.0f, Sg = 0.0f, Sb = 0.0f, Q = 1.0f;
    const int gb = tile + (hi ? 8 : 0);
#pragma unroll
    for (int k = 0; k < 8; ++k) {
      const float p = fminf(pw[k], 0.0f);
      const int gg = gb + k;
      const float al = fminf(op[gg] * __expf(p), 0.99f);
      const float w = al * Q;
      Sr += w * cr[gg];
      Sg += w * cg[gg];
      Sb += w * cb[gg];
      Q *= (1.0f - al);
    }

    // Stitch the two 8-Gaussian segments (lanes 0-15 own the result).
    const float Sr2 = __shfl_xor(Sr, 16);
    const float Sg2 = __shfl_xor(Sg, 16);
    const float Sb2 = __shfl_xor(Sb, 16);
    const float Q2 = __shfl_xor(Q, 16);
    accR += T * (Sr + Q * Sr2);
    accG += T * (Sg + Q * Sg2);
    accB += T * (Sb + Q * Sb2);
    T *= Q * Q2;
  }

  if (!hi) {
    const int y = y0 + lo;
    const int o = (y * GS_W + x) * 3;       // out is [H][W][3]
    const int bo = (x * GS_H + y) * 3;      // bg  is [W][H][3]
    out[o + 0] = accR + bg[bo + 0] * T;
    out[o + 1] = accG + bg[bo + 1] * T;
    out[o + 2] = accB + bg[bo + 2] * T;
  }
}

// ---------------------------------------------------------------------------
extern "C" void kernel_launch(void* const* d_in, const int* in_sizes, int n_in,
                              void* d_out, int out_size, void* d_ws,
                              size_t ws_size, hipStream_t stream) {
  const float* means2D = (const float*)d_in[0];
  const float* depths = (const float*)d_in[1];
  const float* cov2D = (const float*)d_in[2];
  const float* color = (const float*)d_in[3];
  const float* opacity = (const float*)d_in[4];
  const float* bg = (const float*)d_in[5];
  float* ws = (float*)d_ws;  // needs 9*1024*4 = 36 KB
  float* out = (float*)d_out;

  gs_prep<<<1, GS_N, 0, stream>>>(means2D, depths, cov2D, color, opacity, ws);
  gs_render<<<512, 256, 0, stream>>>(ws, bg, out);
}